// InferGraph_1090921693341
// MI455X (gfx1250) — compile-verified
//
#include <hip/hip_runtime.h>
#include <hip/hip_bf16.h>
#include <math.h>

typedef float v2f __attribute__((ext_vector_type(2)));
typedef float v8f __attribute__((ext_vector_type(8)));
typedef int   avi4 __attribute__((vector_size(16)));   // matches builtin param

#define GN 30000
#define GE 240000
#define GD 128
#define GL 6
#define NEG_SLOPE 0.2f
#define LN_EPS 1e-5f
#define PITCH 132   // 128 + 4: stride%64==4 -> 16 rows hit 16 distinct LDS banks

#if defined(__HIP_DEVICE_COMPILE__) && \
    __has_builtin(__builtin_amdgcn_global_load_async_to_lds_b128)
#define HAVE_ASYNC_LDS 1
#else
#define HAVE_ASYNC_LDS 0
#endif

__device__ __forceinline__ float leaky(float v) {
    return v > 0.0f ? v : NEG_SLOPE * v;
}

__device__ __forceinline__ void atomicMaxF32(float* addr, float v) {
    // monotonic bit-pattern trick; m initialized to -inf
    if (v >= 0.0f) atomicMax((int*)addr, __float_as_int(v));
    else           atomicMin((unsigned int*)addr, __float_as_uint(v));
}

// ---------------------------------------------------------------------------
// h[n,k] = emb[x[n], k]
__global__ __launch_bounds__(256)
void gather_kernel(const int* __restrict__ x, const float* __restrict__ emb,
                   float* __restrict__ h, int total) {
    int idx = blockIdx.x * 256 + threadIdx.x;
    if (idx >= total) return;
    int n = idx >> 7, k = idx & 127;
    h[idx] = emb[x[n] * GD + k];
}

// ---------------------------------------------------------------------------
// Wt[c,k] = W[k,c] for a 128x128 matrix (run once per weight; makes the WMMA
// B-fragment a single contiguous b64 load)
__global__ __launch_bounds__(256)
void transpose128(const float* __restrict__ W, float* __restrict__ Wt) {
    int idx = blockIdx.x * 256 + threadIdx.x;   // 16384 elements
    int k = idx >> 7, c = idx & 127;
    Wt[c * GD + k] = W[k * GD + c];
}

// ---------------------------------------------------------------------------
// WMMA f32 GEMM: Y0 = X@W0 + b0 ; if DUAL also Y1 = X@W1 + b1.
// W0t/W1t are pre-TRANSPOSED weights: Wt[c*128 + k] = W[k,c].
// Block = 256 threads = 8 waves; block covers 128 rows; wave w owns rows
// [w*16, w*16+16) and loops over the 8 column tiles.  X tile staged in LDS
// once (async-to-LDS on full tiles); A-fragment shared between both products.
template <bool DUAL>
__global__ __launch_bounds__(256)
void gemm_wmma(const float* __restrict__ X,
               const float* __restrict__ W0t, const float* __restrict__ b0,
               float* __restrict__ Y0,
               const float* __restrict__ W1t, const float* __restrict__ b1,
               float* __restrict__ Y1,
               int nrows) {
    __shared__ float ldsX[128 * PITCH];
    const int tid  = threadIdx.x;
    const int wave = tid >> 5;
    const int lane = tid & 31;
    const int row0 = blockIdx.x * 128;
    const bool full = (row0 + 128) <= nrows;   // uniform across block

    if (full) {
#if HAVE_ASYNC_LDS
        // ASYNCcnt-tracked DMA: 16B per lane per issue, no VGPR staging
        for (int e = tid; e < 128 * 32; e += 256) {
            int r  = e >> 5;
            int c4 = (e & 31) << 2;
            const float* g = X + (row0 + r) * GD + c4;
            float* l = &ldsX[r * PITCH + c4];
            __builtin_amdgcn_global_load_async_to_lds_b128(
                (avi4*)g, (avi4*)l, 0, 0);
        }
#if __has_builtin(__builtin_amdgcn_s_wait_asynccnt)
        __builtin_amdgcn_s_wait_asynccnt(0);
#else
        asm volatile("s_wait_asynccnt 0x0" ::: "memory");
#endif
#else
        for (int e = tid; e < 128 * 128; e += 256) {
            int r = e >> 7, c = e & 127;
            ldsX[r * PITCH + c] = X[(row0 + r) * GD + c];
        }
#endif
    } else {
        for (int e = tid; e < 128 * 128; e += 256) {
            int r = e >> 7, c = e & 127;
            int gr = row0 + r;
            ldsX[r * PITCH + c] = (gr < nrows) ? X[gr * GD + c] : 0.0f;
        }
    }
    __syncthreads();

    const int mrow = lane & 15;            // A row / C column-lane
    const int khi  = (lane >> 4) << 1;     // k offset 0 or 2 per half-wave
    const int crow = (lane >> 4) << 3;     // C: +8 rows for upper half-wave
    const float* lrow = &ldsX[(wave * 16 + mrow) * PITCH];

    for (int nt = 0; nt < 8; ++nt) {
        const int col = nt * 16 + (lane & 15);
        const float* wc0 = W0t + col * GD;
        const float* wc1 = DUAL ? (W1t + col * GD) : nullptr;
        v8f acc0 = {0.f, 0.f, 0.f, 0.f, 0.f, 0.f, 0.f, 0.f};
        v8f acc1 = acc0;
        #pragma unroll 4
        for (int k0 = 0; k0 < GD; k0 += 4) {
            const int kk = k0 + khi;                     // even -> 8B aligned
            const float2 af = *reinterpret_cast<const float2*>(lrow + kk);
            v2f a; a.x = af.x; a.y = af.y;
            const float2 bf = *reinterpret_cast<const float2*>(wc0 + kk);
            v2f b; b.x = bf.x; b.y = bf.y;
            acc0 = __builtin_amdgcn_wmma_f32_16x16x4_f32(
                false, a, false, b, (short)0, acc0, false, false);
            if (DUAL) {
                const float2 cf = *reinterpret_cast<const float2*>(wc1 + kk);
                v2f c; c.x = cf.x; c.y = cf.y;
                acc1 = __builtin_amdgcn_wmma_f32_16x16x4_f32(
                    false, a, false, c, (short)0, acc1, false, false);
            }
        }
        const float bias0 = b0[col];
        const float bias1 = DUAL ? b1[col] : 0.0f;
        const int rbase = row0 + wave * 16 + crow;
        if (full) {                         // uniform: straight-line stores
            #pragma unroll
            for (int j = 0; j < 8; ++j) {
                Y0[(rbase + j) * GD + col] = acc0[j] + bias0;
                if (DUAL) Y1[(rbase + j) * GD + col] = acc1[j] + bias1;
            }
        } else {
            #pragma unroll
            for (int j = 0; j < 8; ++j) {
                if (rbase + j < nrows) {
                    Y0[(rbase + j) * GD + col] = acc0[j] + bias0;
                    if (DUAL) Y1[(rbase + j) * GD + col] = acc1[j] + bias1;
                }
            }
        }
    }
}

// ---------------------------------------------------------------------------
// per-layer scratch reset
__global__ __launch_bounds__(256)
void reset_layer(float* __restrict__ accum, float* __restrict__ denom,
                 float* __restrict__ mbuf, float* __restrict__ scal,
                 int total, int nnodes) {
    int idx = blockIdx.x * 256 + threadIdx.x;
    if (idx < total) accum[idx] = 0.0f;
    if (idx < nnodes) { denom[idx] = 0.0f; mbuf[idx] = __int_as_float(0xff800000); }
    if (idx < 2) scal[idx] = 0.0f;
}

// ---------------------------------------------------------------------------
// one wave32 per edge: score = att . leakyrelu(xl[src] + xr[dst]); seg-max
__global__ __launch_bounds__(256)
void edge_score_max(const float* __restrict__ xl, const float* __restrict__ xr,
                    const float* __restrict__ att, const int* __restrict__ ei,
                    float* __restrict__ score, float* __restrict__ mbuf,
                    int nE, int nE2) {
    int e = blockIdx.x * 8 + (threadIdx.x >> 5);
    if (e >= nE2) return;
    int lane = threadIdx.x & 31;
    int s, d;
    if (e < nE) { s = ei[e]; d = ei[nE + e]; } else { s = e - nE; d = s; }

    const float4 lv = reinterpret_cast<const float4*>(xl + s * GD)[lane];
    const float4 rv = reinterpret_cast<const float4*>(xr + d * GD)[lane];
    const float4 av = reinterpret_cast<const float4*>(att)[lane];
    float p = av.x * leaky(lv.x + rv.x) + av.y * leaky(lv.y + rv.y)
            + av.z * leaky(lv.z + rv.z) + av.w * leaky(lv.w + rv.w);
    #pragma unroll
    for (int off = 16; off > 0; off >>= 1) p += __shfl_xor(p, off, 32);
    if (lane == 0) {
        score[e] = p;
        atomicMaxF32(mbuf + d, p);
    }
}

// ---------------------------------------------------------------------------
// ex = exp(score - m[dst]); denom[dst] += ex; accum[dst,:] += ex * xl[src,:]
__global__ __launch_bounds__(256)
void edge_exp_scatter(const float* __restrict__ xl, const float* __restrict__ score,
                      const float* __restrict__ mbuf, const int* __restrict__ ei,
                      float* __restrict__ denom, float* __restrict__ accum,
                      int nE, int nE2) {
    int e = blockIdx.x * 8 + (threadIdx.x >> 5);
    if (e >= nE2) return;
    int lane = threadIdx.x & 31;
    int s, d;
    if (e < nE) { s = ei[e]; d = ei[nE + e]; } else { s = e - nE; d = s; }

    float ex = __expf(score[e] - mbuf[d]);
    if (lane == 0) atomicAdd(denom + d, ex);
    const float4 lv = reinterpret_cast<const float4*>(xl + s * GD)[lane];
    float* ap = accum + d * GD + lane * 4;
    atomicAdd(ap + 0, ex * lv.x);
    atomicAdd(ap + 1, ex * lv.y);
    atomicAdd(ap + 2, ex * lv.z);
    atomicAdd(ap + 3, ex * lv.w);
}

// ---------------------------------------------------------------------------
// xg = accum/denom + conv_bias ; fused global sum/sumsq for graph-LayerNorm
__global__ __launch_bounds__(256)
void node_finalize(const float* __restrict__ denom, const float* __restrict__ cbias,
                   float* __restrict__ xg, float* __restrict__ scal,
                   int total, int do_stats) {
    __shared__ float wsum[8], wsq[8];
    float ls = 0.0f, lq = 0.0f;
    for (int idx = blockIdx.x * 256 + threadIdx.x; idx < total;
         idx += gridDim.x * 256) {
        int n = idx >> 7, k = idx & 127;
        float v = xg[idx] / denom[n] + cbias[k];   // denom>0 (self-loops)
        xg[idx] = v;
        ls += v; lq += v * v;
    }
    if (!do_stats) return;   // uniform branch
    #pragma unroll
    for (int off = 16; off > 0; off >>= 1) {
        ls += __shfl_xor(ls, off, 32);
        lq += __shfl_xor(lq, off, 32);
    }
    int lane = threadIdx.x & 31, wave = threadIdx.x >> 5;
    if (lane == 0) { wsum[wave] = ls; wsq[wave] = lq; }
    __syncthreads();
    if (threadIdx.x == 0) {
        float s = 0.0f, q = 0.0f;
        #pragma unroll
        for (int w = 0; w < 8; ++w) { s += wsum[w]; q += wsq[w]; }
        atomicAdd(scal + 0, s);
        atomicAdd(scal + 1, q);
    }
}

// ---------------------------------------------------------------------------
__global__ __launch_bounds__(256)
void ln_relu(float* __restrict__ xg, const float* __restrict__ scal,
             const float* __restrict__ g, const float* __restrict__ b, int total) {
    int idx = blockIdx.x * 256 + threadIdx.x;
    if (idx >= total) return;
    float inv = 1.0f / (float)total;
    float mu  = scal[0] * inv;
    float var = scal[1] * inv - mu * mu;
    float rs  = rsqrtf(var + LN_EPS);
    int k = idx & 127;
    float v = (xg[idx] - mu) * rs * g[k] + b[k];
    xg[idx] = v > 0.0f ? v : 0.0f;
}

__global__ __launch_bounds__(256)
void resid_add(const float* __restrict__ h, const float* __restrict__ xg,
               float* __restrict__ t, int total) {
    int idx = blockIdx.x * 256 + threadIdx.x;
    if (idx >= total) return;
    t[idx] = h[idx] + xg[idx];
}

// one block per column: column sum / sumsq over N rows (no atomics)
__global__ __launch_bounds__(256)
void col_reduce(const float* __restrict__ out, float* __restrict__ colsum,
                float* __restrict__ colsq, int nrows) {
    __shared__ float ws[8], wq[8];
    int c = blockIdx.x;
    float s = 0.0f, q = 0.0f;
    for (int r = threadIdx.x; r < nrows; r += 256) {
        float v = out[r * GD + c];
        s += v; q += v * v;
    }
    #pragma unroll
    for (int off = 16; off > 0; off >>= 1) {
        s += __shfl_xor(s, off, 32);
        q += __shfl_xor(q, off, 32);
    }
    int lane = threadIdx.x & 31, wave = threadIdx.x >> 5;
    if (lane == 0) { ws[wave] = s; wq[wave] = q; }
    __syncthreads();
    if (threadIdx.x == 0) {
        float ts = 0.0f, tq = 0.0f;
        #pragma unroll
        for (int w = 0; w < 8; ++w) { ts += ws[w]; tq += wq[w]; }
        colsum[c] = ts; colsq[c] = tq;
    }
}

__global__ __launch_bounds__(256)
void bn_apply(float* __restrict__ out, const float* __restrict__ colsum,
              const float* __restrict__ colsq, const float* __restrict__ g,
              const float* __restrict__ b, int nrows, int total) {
    int idx = blockIdx.x * 256 + threadIdx.x;
    if (idx >= total) return;
    int c = idx & 127;
    float invn = 1.0f / (float)nrows;
    float mu  = colsum[c] * invn;
    float var = colsq[c] * invn - mu * mu;
    out[idx] = (out[idx] - mu) * rsqrtf(var + LN_EPS) * g[c] + b[c];
}

// ---------------------------------------------------------------------------
extern "C" void kernel_launch(void* const* d_in, const int* in_sizes, int n_in,
                              void* d_out, int out_size, void* d_ws, size_t ws_size,
                              hipStream_t stream) {
    const int*   x    = (const int*)d_in[0];
    const int*   ei   = (const int*)d_in[1];
    const float* emb  = (const float*)d_in[2];
    const float* Wl   = (const float*)d_in[3];
    const float* bl   = (const float*)d_in[4];
    const float* Wr   = (const float*)d_in[5];
    const float* br   = (const float*)d_in[6];
    const float* att  = (const float*)d_in[7];
    const float* cb   = (const float*)d_in[8];
    const float* lng  = (const float*)d_in[9];
    const float* lnb  = (const float*)d_in[10];
    const float* linW = (const float*)d_in[11];
    const float* linb = (const float*)d_in[12];
    const float* bng  = (const float*)d_in[13];
    const float* bnb  = (const float*)d_in[14];
    float* out = (float*)d_out;

    const int N = GN, E = GE, D = GD;
    const int E2 = E + N;
    const int ND = N * D;
    const int DD = D * D;

    float* ws    = (float*)d_ws;
    float* h     = ws;  ws += ND;
    float* xgA   = ws;  ws += ND;
    float* xgB   = ws;  ws += ND;
    float* xl    = ws;  ws += ND;
    float* xr    = ws;  ws += ND;
    float* score = ws;  ws += E2;
    float* mbuf  = ws;  ws += N;
    float* denom = ws;  ws += N;
    float* csum  = ws;  ws += D;
    float* csq   = ws;  ws += D;
    float* scal  = ws;  ws += 2;
    float* wtl   = ws;  ws += GL * DD;   // transposed Wl
    float* wtr   = ws;  ws += GL * DD;   // transposed Wr
    float* wtlin = ws;  ws += DD;        // transposed lin_W

    const dim3 blk(256);
    const int gridND     = (ND + 255) / 256;
    const int gridDD     = DD / 256;     // 64
    const int gemmBlocks = (N + 127) / 128;
    const int edgeBlocks = (E2 + 7) / 8;

    // one-time weight transposes (cheap: 13 x 64KB)
    for (int i = 0; i < GL; ++i) {
        transpose128<<<gridDD, blk, 0, stream>>>(Wl + i * DD, wtl + i * DD);
        transpose128<<<gridDD, blk, 0, stream>>>(Wr + i * DD, wtr + i * DD);
    }
    transpose128<<<gridDD, blk, 0, stream>>>(linW, wtlin);

    gather_kernel<<<gridND, blk, 0, stream>>>(x, emb, h, ND);

    float* xin = h;
    for (int i = 0; i < GL; ++i) {
        float* xout = (i & 1) ? xgB : xgA;
        gemm_wmma<true><<<gemmBlocks, blk, 0, stream>>>(
            xin, wtl + i * DD, bl + i * D, xl,
                 wtr + i * DD, br + i * D, xr, N);
        reset_layer<<<gridND, blk, 0, stream>>>(xout, denom, mbuf, scal, ND, N);
        edge_score_max<<<edgeBlocks, blk, 0, stream>>>(
            xl, xr, att + i * D, ei, score, mbuf, E, E2);
        edge_exp_scatter<<<edgeBlocks, blk, 0, stream>>>(
            xl, score, mbuf, ei, denom, xout, E, E2);
        node_finalize<<<480, blk, 0, stream>>>(
            denom, cb + i * D, xout, scal, ND, (i < GL - 1) ? 1 : 0);
        if (i < GL - 1) {
            ln_relu<<<gridND, blk, 0, stream>>>(
                xout, scal, lng + i * D, lnb + i * D, ND);
        }
        xin = xout;
    }

    // out = (h + xg) @ lin_W + lin_b, then train-mode BatchNorm1d
    resid_add<<<gridND, blk, 0, stream>>>(h, xin, xl, ND);
    gemm_wmma<false><<<gemmBlocks, blk, 0, stream>>>(
        xl, wtlin, linb, out, nullptr, nullptr, nullptr, N);
    col_reduce<<<D, blk, 0, stream>>>(out, csum, csq, N);
    bn_apply<<<gridND, blk, 0, stream>>>(out, csum, csq, bng, bnb, N, ND);
}